// Channel_Embedding_38783554683258
// MI455X (gfx1250) — compile-verified
//
#include <hip/hip_runtime.h>

#define G   8
#define DIM 32
#define E   8
#define OC  10
#define BB  64
#define L   4096
#define LP  4094   // L - 2

typedef __attribute__((ext_vector_type(2))) float v2f;
typedef __attribute__((ext_vector_type(8))) float v8f;

#define COMBINE_SIZE (BB * G * OC * LP)
#define LOSS_OFF     COMBINE_SIZE
#define GATES_OFF    (COMBINE_SIZE + 1)

// ---------------------------------------------------------------------------
// Kernel 1: noisy_top_k_gating (eval path). One wave per (b,g).
// ---------------------------------------------------------------------------
__global__ __launch_bounds__(32) void gating_kernel(
    const float* __restrict__ x, const float* __restrict__ wg,
    float* __restrict__ gates_ws, float* __restrict__ out) {
  int bg = blockIdx.x;
  int b = bg / G, g = bg % G;
  int lane = threadIdx.x;

  float part[E];
#pragma unroll
  for (int e = 0; e < E; e++) part[e] = 0.f;

  // gin[d*5+t] = x[b, g*32+d, L-6+t], t=0..4
  for (int idx = lane; idx < DIM * 5; idx += 32) {
    int d = idx / 5, t = idx % 5;
    float xv = x[((size_t)bg * DIM + d) * L + (L - 6) + t];
    const float* w = wg + ((size_t)g * DIM * 5 + idx) * E;
#pragma unroll
    for (int e = 0; e < E; e++) part[e] += xv * w[e];
  }
#pragma unroll
  for (int off = 16; off > 0; off >>= 1)
#pragma unroll
    for (int e = 0; e < E; e++) part[e] += __shfl_down(part[e], off, 32);

  if (lane == 0) {
    float mx = part[0];
#pragma unroll
    for (int e = 1; e < E; e++) mx = fmaxf(mx, part[e]);
    float p[E], s = 0.f;
#pragma unroll
    for (int e = 0; e < E; e++) { p[e] = __expf(part[e] - mx); s += p[e]; }
#pragma unroll
    for (int e = 0; e < E; e++) p[e] /= s;
    // top-2 (first occurrence on ties, like lax.top_k)
    int i1 = 0;
#pragma unroll
    for (int e = 1; e < E; e++) if (p[e] > p[i1]) i1 = e;
    int i2 = (i1 == 0) ? 1 : 0;
#pragma unroll
    for (int e = 0; e < E; e++) if (e != i1 && p[e] > p[i2]) i2 = e;
    float denom = p[i1] + p[i2] + 1e-6f;
    float gts[E];
#pragma unroll
    for (int e = 0; e < E; e++) gts[e] = 0.f;
    gts[i1] = p[i1] / denom;
    gts[i2] = p[i2] / denom;
#pragma unroll
    for (int e = 0; e < E; e++) {
      gates_ws[(size_t)bg * E + e] = gts[e];
      out[GATES_OFF + ((size_t)b * E + e) * G + g] = gts[e];  // [B,E,G]
    }
  }
}

// ---------------------------------------------------------------------------
// Kernel 2: cv_squared load-balance loss. One block.
// ---------------------------------------------------------------------------
__global__ __launch_bounds__(128) void loss_kernel(
    const float* __restrict__ gates_ws, float* __restrict__ out) {
  __shared__ float imp_s[G * E], load_s[G * E], cv_s[G];
  int tid = threadIdx.x;
  if (tid < G * E) {
    int g = tid >> 3, e = tid & 7;
    float imp = 0.f, ld = 0.f;
    for (int b = 0; b < BB; b++) {
      float v = gates_ws[((size_t)b * G + g) * E + e];
      imp += v;
      ld += (v > 0.f) ? 1.f : 0.f;
    }
    imp_s[tid] = imp;
    load_s[tid] = ld;
  }
  __syncthreads();
  if (tid < G) {
    float loss = 0.f;
    for (int pass = 0; pass < 2; pass++) {
      const float* v = pass ? load_s : imp_s;
      float m = 0.f;
#pragma unroll
      for (int e = 0; e < E; e++) m += v[tid * E + e];
      m /= (float)E;
      float var = 0.f;
#pragma unroll
      for (int e = 0; e < E; e++) { float d = v[tid * E + e] - m; var += d * d; }
      var /= (float)(E - 1);  // ddof=1
      loss += var / (m * m + 1e-10f);
    }
    cv_s[tid] = loss;
  }
  __syncthreads();
  if (tid == 0) {
    float s = 0.f;
#pragma unroll
    for (int g = 0; g < G; g++) s += cv_s[g];
    out[LOSS_OFF] = 0.01f * s;
  }
}

// ---------------------------------------------------------------------------
// Kernel 3: fused conv1(tanh) + gate-collapsed conv2 via f32 WMMA.
// Block = one (b,g) x 256 output columns; 4 waves x 4 tiles of 16 cols.
// LDS layouts are channel-PAIR interleaved so every WMMA B-fragment is one
// aligned ds_load_b64 at an immediate offset from a single base VGPR.
// ---------------------------------------------------------------------------
#define XT      256
#define PPITCH  544   // floats per pair-row: 544%64==32 -> split-lane reads conflict-free
#define HP      40    // h pair pitch: (4*40)%64==32 -> conflict-free h stores
#define NTPW    4     // tiles per wave

__global__ __launch_bounds__(128) void moe_conv_kernel(
    const float* __restrict__ x, const float* __restrict__ w1,
    const float* __restrict__ b1, const float* __restrict__ w2,
    const float* __restrict__ b2, const float* __restrict__ gates_ws,
    float* __restrict__ out) {
  __shared__ float xs[(DIM / 2) * PPITCH];   // xp[p][c*2 + (d&1)], p = d/2
  __shared__ float w1s[OC * DIM * 3];
  __shared__ float w2s[16 * 16];
  __shared__ float hs[4 * 8 * HP];           // per-wave hp[c/2][2n + (c&1)]
  __shared__ float b1s[16], b2s[16];

  int blk = blockIdx.x;
  int j   = blk & 15;      // column chunk
  int bg  = blk >> 4;      // b*G+g
  int g   = bg & 7;
  int tid = threadIdx.x;
  int lane = tid & 31, wave = tid >> 5;
  int col0 = j * XT;

  const float* xg = x + (size_t)bg * DIM * L;

  // ---- stage x tile, pair-interleaved: xs[p*PPITCH + c*2 + par] ----
  for (int idx = tid; idx < (DIM / 2) * 65; idx += 128) {
    int p = idx / 65, c4 = idx % 65;
    int gc = col0 + c4 * 4;
    const float* r0p = xg + (size_t)(2 * p) * L;
    const float* r1p = xg + (size_t)(2 * p + 1) * L;
    float4 r0, r1;
    if (gc + 3 < L) {
      r0 = *(const float4*)(r0p + gc);
      r1 = *(const float4*)(r1p + gc);
    } else {
      float a0 = (gc + 0 < L) ? r0p[gc + 0] : 0.f;
      float a1v = (gc + 1 < L) ? r0p[gc + 1] : 0.f;
      float a2v = (gc + 2 < L) ? r0p[gc + 2] : 0.f;
      float a3 = (gc + 3 < L) ? r0p[gc + 3] : 0.f;
      float c0v = (gc + 0 < L) ? r1p[gc + 0] : 0.f;
      float c1 = (gc + 1 < L) ? r1p[gc + 1] : 0.f;
      float c2 = (gc + 2 < L) ? r1p[gc + 2] : 0.f;
      float c3 = (gc + 3 < L) ? r1p[gc + 3] : 0.f;
      r0 = make_float4(a0, a1v, a2v, a3);
      r1 = make_float4(c0v, c1, c2, c3);
    }
    float* dst = xs + p * PPITCH + c4 * 8;   // 8 contiguous floats
    ((float2*)dst)[0] = make_float2(r0.x, r1.x);
    ((float2*)dst)[1] = make_float2(r0.y, r1.y);
    ((float2*)dst)[2] = make_float2(r0.z, r1.z);
    ((float2*)dst)[3] = make_float2(r0.w, r1.w);
  }

  // ---- conv1 weights for this group ----
  for (int idx = tid; idx < OC * DIM * 3; idx += 128)
    w1s[idx] = w1[(size_t)g * OC * DIM * 3 + idx];

  // ---- gate-collapsed 10x10 conv2 weight (zero-padded to 16x16) ----
  const float* gates = gates_ws + (size_t)bg * E;
  for (int idx = tid; idx < 256; idx += 128) {
    int r = idx >> 4, c = idx & 15;
    float v = 0.f;
    if (r < OC && c < OC) {
#pragma unroll
      for (int e = 0; e < E; e++)
        v += gates[e] * w2[((size_t)(g * OC + r) * E + e) * OC + c];
    }
    w2s[idx] = v;
  }
  if (tid < 16) {
    b1s[tid] = (tid < OC) ? b1[g * OC + tid] : 0.f;
    float v = 0.f;
    if (tid < OC) {
#pragma unroll
      for (int e = 0; e < E; e++)
        v += gates[e] * b2[(size_t)(g * OC + tid) * E + e];
    }
    b2s[tid] = v;
  }
  __syncthreads();

  // ---- per-lane fragment geometry (ISA 7.12.2 layouts) ----
  int m      = lane & 15;
  int koff   = (lane < 16) ? 0 : 2;   // A/B 16x4 split-lane K offset (even!)
  int koffp  = koff >> 1;             // pair index offset
  int rowoff = (lane < 16) ? 0 : 8;   // C/D row offset
  int n      = lane & 15;             // C/D / B column

  // A fragments, held in registers across all tiles
  v2f a1[24];
#pragma unroll
  for (int kk = 0; kk < 3; kk++)
#pragma unroll
    for (int s = 0; s < 8; s++) {
      int d0 = 4 * s + koff;
      v2f a; a.x = 0.f; a.y = 0.f;
      if (m < OC) {
        a.x = w1s[(m * DIM + d0) * 3 + kk];
        a.y = w1s[(m * DIM + d0 + 1) * 3 + kk];
      }
      a1[kk * 8 + s] = a;
    }
  v2f a2[4];
#pragma unroll
  for (int s = 0; s < 4; s++) {
    int c0 = 4 * s + koff;
    v2f a; a.x = w2s[m * 16 + c0]; a.y = w2s[m * 16 + c0 + 1];
    a2[s] = a;
  }
  float bA[8], bB2[8];
#pragma unroll
  for (int r = 0; r < 8; r++) { bA[r] = b1s[r + rowoff]; bB2[r] = b2s[r + rowoff]; }

  float* hw  = hs + wave * 8 * HP;          // this wave's h scratch
  float* hwp = hw + (rowoff >> 1) * HP;     // store base (pairs 0..3 or 4..7)
  const float* hb = hw + koffp * HP + 2 * n;  // load base

  for (int it = 0; it < NTPW; it++) {
    int tcol = (wave * NTPW + it) * 16;
    // single base VGPR; all 24 fragment loads use immediate offsets
    const float* xb = xs + koffp * PPITCH + (tcol + n) * 2;

    // ---- stage 1: conv1 as 3 shifted GEMMs, K=32 each (24 WMMA) ----
    v8f acc = {0.f, 0.f, 0.f, 0.f, 0.f, 0.f, 0.f, 0.f};
#pragma unroll
    for (int kk = 0; kk < 3; kk++) {
#pragma unroll
      for (int s = 0; s < 8; s++) {
        v2f bf = *(const v2f*)(xb + s * 2 * PPITCH + kk * 2);  // ds_load_b64
        acc = __builtin_amdgcn_wmma_f32_16x16x4_f32(
            false, a1[kk * 8 + s], false, bf, (short)0, acc, false, false);
      }
    }

    // ---- h = tanh(acc + b1), park in LDS pair-interleaved (b64 stores) ----
#pragma unroll
    for (int k = 0; k < 4; k++) {
      float2 hv = make_float2(tanhf(acc[2 * k] + bA[2 * k]),
                              tanhf(acc[2 * k + 1] + bA[2 * k + 1]));
      *(float2*)(hwp + k * HP + 2 * n) = hv;
    }
    __syncthreads();  // order LDS h store -> B-layout reload

    // ---- stage 2: 10x10 gate-collapsed conv2, K=16 (4 WMMA) ----
    v8f acc2 = {0.f, 0.f, 0.f, 0.f, 0.f, 0.f, 0.f, 0.f};
#pragma unroll
    for (int s = 0; s < 4; s++) {
      v2f bf = *(const v2f*)(hb + s * 2 * HP);                 // ds_load_b64
      acc2 = __builtin_amdgcn_wmma_f32_16x16x4_f32(
          false, a2[s], false, bf, (short)0, acc2, false, false);
    }
    __syncthreads();  // WAR: protect hw before next tile's stores

    // ---- epilogue: bias + store [B,G,OC,LP] ----
    int gl = col0 + tcol + n;
    if (gl < LP) {
#pragma unroll
      for (int r = 0; r < 8; r++) {
        int mm = r + rowoff;
        if (mm < OC)
          out[((size_t)bg * OC + mm) * LP + gl] = acc2[r] + bB2[r];
      }
    }
  }
}

// ---------------------------------------------------------------------------
extern "C" void kernel_launch(void* const* d_in, const int* in_sizes, int n_in,
                              void* d_out, int out_size, void* d_ws, size_t ws_size,
                              hipStream_t stream) {
  const float* x  = (const float*)d_in[0];
  const float* w1 = (const float*)d_in[1];
  const float* b1 = (const float*)d_in[2];
  const float* w2 = (const float*)d_in[3];
  const float* b2 = (const float*)d_in[4];
  const float* wg = (const float*)d_in[5];
  float* out = (float*)d_out;
  float* gates_ws = (float*)d_ws;  // B*G*E floats = 16 KB

  gating_kernel<<<BB * G, 32, 0, stream>>>(x, wg, gates_ws, out);
  loss_kernel<<<1, 128, 0, stream>>>(gates_ws, out);
  moe_conv_kernel<<<BB * G * 16, 128, 0, stream>>>(x, w1, b1, w2, b2,
                                                   gates_ws, out);
}